// Net_2894807957580
// MI455X (gfx1250) — compile-verified
//
#include <hip/hip_runtime.h>
#include <hip/hip_bf16.h>

// ---------------------------------------------------------------------------
// GCN (2x GCNConv + mean-pool + linear + softmax) for MI455X / gfx1250.
//
// Bandwidth analysis: edge scatter/gather dominates (~3 GB @ 23.3 TB/s ->
// ~130us floor). GEMMs are tiny (1.2 GFLOP) but are mapped onto
// V_WMMA_F32_16X16X4_F32 (full f32 precision, wave32, 16x16 tiles).
// ---------------------------------------------------------------------------

typedef __attribute__((ext_vector_type(2))) float v2f;
typedef __attribute__((ext_vector_type(8))) float v8f;

// ---------------- elementwise / utility kernels ----------------

__global__ void k_fill(float* __restrict__ p, float v, int n) {
    int i = blockIdx.x * blockDim.x + threadIdx.x;
    if (i < n) p[i] = v;
}

__global__ void k_deg_edges(const int* __restrict__ row, float* __restrict__ deg, int E) {
    int e = blockIdx.x * blockDim.x + threadIdx.x;
    if (e < E) atomicAdd(&deg[row[e]], 1.0f);
}

__global__ void k_rsqrt(float* __restrict__ d, int n) {
    int i = blockIdx.x * blockDim.x + threadIdx.x;
    if (i < n) d[i] = rsqrtf(d[i]);
}

__global__ void k_norm(const int* __restrict__ row, const int* __restrict__ col,
                       const float* __restrict__ dis, float* __restrict__ norm, int E) {
    int e = blockIdx.x * blockDim.x + threadIdx.x;
    if (e < E) norm[e] = dis[row[e]] * dis[col[e]];
}

__global__ void k_relu(float* __restrict__ p, int n) {
    int i = blockIdx.x * blockDim.x + threadIdx.x;
    if (i < n) p[i] = fmaxf(p[i], 0.0f);
}

// ---------------- WMMA linear layer: out = X @ W^T + b ----------------
// X: [N, K] row-major, W: [FOUT, K] row-major (torch Linear weight), out: [N, FOUT].
// One wave computes a 16-row tile of output; FOUT/16 N-tiles per wave.
// f32 WMMA fragment layout (ISA 7.12.2, 32-bit A 16x4):
//   A: lane L holds row M=L%16; VGPR j holds K = j + 2*(L/16)   -> float2 load
//   B: lane L holds col N=L%16; VGPR j holds K = j + 2*(L/16)   -> float2 load
//   C/D: lane L holds col N=L%16; VGPR v holds row M = v + 8*(L/16)
template <int K, int FOUT>
__global__ void k_lin_wmma(const float* __restrict__ X, const float* __restrict__ W,
                           const float* __restrict__ bias, float* __restrict__ out, int N) {
    int wave  = (blockIdx.x * blockDim.x + threadIdx.x) >> 5;
    int tiles = (N + 15) >> 4;
    if (wave >= tiles) return;          // wave-uniform: EXEC stays all-1s for WMMA
    int lane = threadIdx.x & 31;
    int hi   = lane >> 4;               // 0 or 1
    int l16  = lane & 15;
    int rowIdx = wave * 16 + l16;
    if (rowIdx >= N) rowIdx = N - 1;    // clamp (N % 16 == 0 for this problem)

    const float* aBase = X + (size_t)rowIdx * K + 2 * hi;
#pragma unroll
    for (int nt = 0; nt < FOUT / 16; ++nt) {
        const int n0 = nt * 16;
        const float* bBase = W + (size_t)(n0 + l16) * K + 2 * hi;
        v8f acc = {};
#pragma unroll
        for (int k0 = 0; k0 < K; k0 += 4) {
            v2f a = *(const v2f*)(aBase + k0);
            v2f b = *(const v2f*)(bBase + k0);
            // 8 args: (neg_a, A, neg_b, B, c_mod, C, reuse_a, reuse_b)
            acc = __builtin_amdgcn_wmma_f32_16x16x4_f32(
                false, a, false, b, (short)0, acc, false, false);
        }
        float bv = bias[n0 + l16];
#pragma unroll
        for (int v = 0; v < 8; ++v) {
            int m = wave * 16 + v + 8 * hi;
            if (m < N) out[(size_t)m * FOUT + n0 + l16] = acc[v] + bv;
        }
    }
}

// ---------------- self-loop init: agg[i,:] = h[i,:] * dis[i]^2 ----------------
template <int F>
__global__ void k_selfloop(const float* __restrict__ h, const float* __restrict__ dis,
                           float* __restrict__ agg, int N) {
    int i = blockIdx.x * blockDim.x + threadIdx.x;
    if (i < N * F) {
        float d = dis[i / F];
        agg[i] = h[i] * d * d;
    }
}

// ---------------- edge scatter (the bandwidth-dominant kernels) ----------------
// One wave32 per edge; lane = feature. Coalesced 128B gather + 128B atomic-add.
__global__ void k_scatter32(const int* __restrict__ row, const int* __restrict__ col,
                            const float* __restrict__ norm, const float* __restrict__ h,
                            float* __restrict__ agg, int E) {
    int e = (blockIdx.x * blockDim.x + threadIdx.x) >> 5;
    if (e >= E) return;
    int lane = threadIdx.x & 31;
    int r = row[e], c = col[e];
    float w = norm[e];
    atomicAdd(&agg[(size_t)c * 32 + lane], w * h[(size_t)r * 32 + lane]);
}

__global__ void k_scatter64(const int* __restrict__ row, const int* __restrict__ col,
                            const float* __restrict__ norm, const float* __restrict__ h,
                            float* __restrict__ agg, int E) {
    int e = (blockIdx.x * blockDim.x + threadIdx.x) >> 5;
    if (e >= E) return;
    int lane = threadIdx.x & 31;
    int r = row[e], c = col[e];
    float w = norm[e];
    const float* hp = h + (size_t)r * 64;
    float* ap = agg + (size_t)c * 64;
    atomicAdd(&ap[lane], w * hp[lane]);
    atomicAdd(&ap[lane + 32], w * hp[lane + 32]);
}

// ---------------- pooling ----------------
__global__ void k_counts(const int* __restrict__ batch, float* __restrict__ counts, int N) {
    int i = blockIdx.x * blockDim.x + threadIdx.x;
    if (i < N) atomicAdd(&counts[batch[i]], 1.0f);
}

__global__ void k_pool(const float* __restrict__ h, const int* __restrict__ batch,
                       float* __restrict__ sums, int N) {
    int i = blockIdx.x * blockDim.x + threadIdx.x;
    if (i < N * 64) {
        int node = i >> 6;
        int f = i & 63;
        atomicAdd(&sums[(size_t)batch[node] * 64 + f], h[i]);
    }
}

// ---------------- head: mean, linear [64,10], softmax ----------------
__global__ void k_head(const float* __restrict__ sums, const float* __restrict__ counts,
                       const float* __restrict__ Wf, const float* __restrict__ bf,
                       float* __restrict__ out, int G) {
    int g = blockIdx.x * blockDim.x + threadIdx.x;
    if (g >= G) return;
    float inv = 1.0f / fmaxf(counts[g], 1.0f);
    float logits[10];
#pragma unroll
    for (int c = 0; c < 10; ++c) {
        float acc = bf[c];
        for (int f = 0; f < 64; ++f)
            acc += (sums[(size_t)g * 64 + f] * inv) * Wf[(size_t)c * 64 + f];
        logits[c] = acc;
    }
    float mx = logits[0];
#pragma unroll
    for (int c = 1; c < 10; ++c) mx = fmaxf(mx, logits[c]);
    float s = 0.0f;
#pragma unroll
    for (int c = 0; c < 10; ++c) { float ev = __expf(logits[c] - mx); logits[c] = ev; s += ev; }
    float is = 1.0f / s;
#pragma unroll
    for (int c = 0; c < 10; ++c) out[(size_t)g * 10 + c] = logits[c] * is;
}

// ---------------------------------------------------------------------------

static inline unsigned cdiv(long long n, int b) { return (unsigned)((n + b - 1) / b); }

extern "C" void kernel_launch(void* const* d_in, const int* in_sizes, int n_in,
                              void* d_out, int out_size, void* d_ws, size_t ws_size,
                              hipStream_t stream) {
    const float* x     = (const float*)d_in[0];
    const int*   ei    = (const int*)d_in[1];
    const int*   batch = (const int*)d_in[2];
    const float* W1    = (const float*)d_in[3];
    const float* b1    = (const float*)d_in[4];
    const float* W2    = (const float*)d_in[5];
    const float* b2    = (const float*)d_in[6];
    const float* Wf    = (const float*)d_in[7];
    const float* bf    = (const float*)d_in[8];
    float* out = (float*)d_out;

    const int N = in_sizes[0] / 128;   // 100000
    const int E = in_sizes[1] / 2;     // 3200000
    const int G = 64;
    const int* row = ei;               // edge_index[0] (source)
    const int* col = ei + E;           // edge_index[1] (target)

    // workspace layout (floats)
    float* ws     = (float*)d_ws;
    float* dis    = ws;  ws += N;               // deg -> rsqrt(deg) in place
    float* nrm    = ws;  ws += E;               // per-edge norm
    float* hl1    = ws;  ws += (size_t)N * 32;  // lin1 output
    float* h1     = ws;  ws += (size_t)N * 32;  // agg1 -> relu
    float* hl2    = ws;  ws += (size_t)N * 64;  // lin2 output
    float* h2     = ws;  ws += (size_t)N * 64;  // agg2 -> relu
    float* sums   = ws;  ws += (size_t)G * 64;  // pooled sums
    float* counts = ws;  ws += G;               // contiguous after sums

    const dim3 B(256);

    // degrees (self-loop contributes 1) -> dis = deg^-0.5 -> per-edge norm
    k_fill     <<<cdiv(N, 256), B, 0, stream>>>(dis, 1.0f, N);
    k_deg_edges<<<cdiv(E, 256), B, 0, stream>>>(row, dis, E);
    k_rsqrt    <<<cdiv(N, 256), B, 0, stream>>>(dis, N);
    k_norm     <<<cdiv(E, 256), B, 0, stream>>>(row, col, dis, nrm, E);

    const long long waveThreads = (long long)((N + 15) / 16) * 32;

    // layer 1: lin (WMMA f32) -> self-loop init -> edge scatter -> relu
    k_lin_wmma<128, 32><<<cdiv(waveThreads, 256), B, 0, stream>>>(x, W1, b1, hl1, N);
    k_selfloop<32>     <<<cdiv((long long)N * 32, 256), B, 0, stream>>>(hl1, dis, h1, N);
    k_scatter32        <<<cdiv((long long)E * 32, 256), B, 0, stream>>>(row, col, nrm, hl1, h1, E);
    k_relu             <<<cdiv((long long)N * 32, 256), B, 0, stream>>>(h1, N * 32);

    // layer 2
    k_lin_wmma<32, 64> <<<cdiv(waveThreads, 256), B, 0, stream>>>(h1, W2, b2, hl2, N);
    k_selfloop<64>     <<<cdiv((long long)N * 64, 256), B, 0, stream>>>(hl2, dis, h2, N);
    k_scatter64        <<<cdiv((long long)E * 32, 256), B, 0, stream>>>(row, col, nrm, hl2, h2, E);
    k_relu             <<<cdiv((long long)N * 64, 256), B, 0, stream>>>(h2, N * 64);

    // mean-pool + head (sums and counts are contiguous -> one fill)
    k_fill  <<<cdiv(G * 64 + G, 256), B, 0, stream>>>(sums, 0.0f, G * 64 + G);
    k_counts<<<cdiv(N, 256), B, 0, stream>>>(batch, counts, N);
    k_pool  <<<cdiv((long long)N * 64, 256), B, 0, stream>>>(h2, batch, sums, N);
    k_head  <<<1, 64, 0, stream>>>(sums, counts, Wf, bf, out, G);
}